// GraphAttentionLayer_52475910422917
// MI455X (gfx1250) — compile-verified
//
#include <hip/hip_runtime.h>
#include <math.h>

typedef __attribute__((ext_vector_type(2))) float v2f;
typedef __attribute__((ext_vector_type(8))) float v8f;

#define N_NODES   50000
#define N_EDGES   800000
#define IN_F      256
#define EDGE_F    64
#define NH        8
#define NF        32
#define HF        256   // NH*NF

// ---------------- order-preserving f32 <-> u32 encoding for atomicMax ------
__device__ __forceinline__ unsigned enc_f32(float f) {
  unsigned u = __float_as_uint(f);
  return (u & 0x80000000u) ? ~u : (u | 0x80000000u);
}
__device__ __forceinline__ float dec_f32(unsigned u) {
  unsigned b = (u & 0x80000000u) ? (u ^ 0x80000000u) : ~u;
  return __uint_as_float(b);
}

// ---------------- 1) Wh = h @ W  via V_WMMA_F32_16X16X4_F32 ----------------
// block = 256 threads = 8 waves; block computes rows [16*bx,16*bx+16),
// wave w computes cols [32*w, 32*w+32) (exactly one attention head).
__global__ __launch_bounds__(256)
void gemm_wh_wmma(const float* __restrict__ h, const float* __restrict__ W,
                  float* __restrict__ Wh) {
  const int lane = threadIdx.x & 31;
  const int wave = threadIdx.x >> 5;
  const int m0   = blockIdx.x * 16;
  const int n0   = wave * 32;
  const int l15  = lane & 15;
  const int hi   = lane >> 4;          // 0 for lanes 0-15, 1 for lanes 16-31
  const int khalf = hi * 2;            // A/B hold K={0,1} or K={2,3}

  v8f c0 = {}; v8f c1 = {};
  const float* arow = h + (size_t)(m0 + l15) * IN_F;

  for (int k0 = 0; k0 < IN_F; k0 += 4) {
    // A fragment 16x4: lane holds (M=lane&15, K=khalf..khalf+1) -> float2
    v2f a = *(const v2f*)(arow + k0 + khalf);
    // B fragment 4x16: lane holds (K=khalf..khalf+1, N=lane&15)
    const float* bb = W + (size_t)(k0 + khalf) * HF + n0 + l15;
    v2f b0, b1;
    b0.x = bb[0];        b0.y = bb[HF];
    b1.x = bb[16];       b1.y = bb[HF + 16];
    c0 = __builtin_amdgcn_wmma_f32_16x16x4_f32(false, a, false, b0,
                                               (short)0, c0, false, false);
    c1 = __builtin_amdgcn_wmma_f32_16x16x4_f32(false, a, false, b1,
                                               (short)0, c1, false, false);
  }
  // D layout: VGPR r, lanes 0-15 -> M=r, lanes 16-31 -> M=r+8; N = lane&15
  const int mo = m0 + (hi << 3);
  const int nc = n0 + l15;
  for (int r = 0; r < 8; ++r) {
    Wh[(size_t)(mo + r) * HF + nc]      = c0[r];
    Wh[(size_t)(mo + r) * HF + nc + 16] = c1[r];
  }
}

// ---------------- 2) fold edge weights: V[k,h] = sum_f We_w[k,h*32+f]*a_edge[h,f]
__global__ void fold_we(const float* __restrict__ We_w,
                        const float* __restrict__ a_edge,
                        float* __restrict__ V) {
  int t = threadIdx.x;                 // 512 threads
  int k = t >> 3, hh = t & 7;
  const float* wr = We_w + (size_t)k * HF + hh * NF;
  const float* ae = a_edge + hh * NF;
  float acc = 0.f;
  for (int f = 0; f < NF; ++f) acc += wr[f] * ae[f];
  V[k * NH + hh] = acc;
}

// ---------------- 3) per-node score halves: s_src[n,h], s_dst[n,h] ---------
__global__ __launch_bounds__(256)
void node_scores(const float* __restrict__ Wh, const float* __restrict__ a_src,
                 const float* __restrict__ a_dst,
                 float* __restrict__ ssrc, float* __restrict__ sdst) {
  int t = blockIdx.x * blockDim.x + threadIdx.x;
  if (t >= N_NODES * NH) return;
  int n = t >> 3, hh = t & 7;
  const float* row = Wh + (size_t)n * HF + hh * NF;
  const float* as  = a_src + hh * NF;
  const float* ad  = a_dst + hh * NF;
  float accs = 0.f, accd = 0.f;
  for (int f = 0; f < NF; ++f) { float v = row[f]; accs += v * as[f]; accd += v * ad[f]; }
  ssrc[t] = accs;
  sdst[t] = accd;
}

// ---------------- 4) per-(edge,head) logit + leakyrelu + segment max -------
__global__ __launch_bounds__(256)
void edge_logits(const float* __restrict__ ef, const int* __restrict__ src,
                 const int* __restrict__ dst, const float* __restrict__ V,
                 const float* __restrict__ ssrc, const float* __restrict__ sdst,
                 float* __restrict__ sbuf, unsigned* __restrict__ m) {
  __shared__ float Vl[EDGE_F * NH];
  for (int i = threadIdx.x; i < EDGE_F * NH; i += blockDim.x) Vl[i] = V[i];
  __syncthreads();
  int t = blockIdx.x * blockDim.x + threadIdx.x;
  if (t >= N_EDGES * NH) return;
  int e = t >> 3, hh = t & 7;
  const float4* efr = (const float4*)(ef + (size_t)e * EDGE_F);
  float acc = 0.f;
  for (int q = 0; q < EDGE_F / 4; ++q) {
    float4 v = efr[q];
    int k = q * 4;
    acc += v.x * Vl[(k + 0) * NH + hh] + v.y * Vl[(k + 1) * NH + hh]
         + v.z * Vl[(k + 2) * NH + hh] + v.w * Vl[(k + 3) * NH + hh];
  }
  int s = src[e], d = dst[e];
  float sv = ssrc[s * NH + hh] + sdst[d * NH + hh] + acc;
  sv = sv > 0.f ? sv : 0.01f * sv;           // LeakyReLU(0.01)
  sbuf[t] = sv;
  atomicMax(&m[d * NH + hh], enc_f32(sv));
}

// ---------------- 5) ex = exp(s - m[dst]);  den[dst] += ex -----------------
__global__ __launch_bounds__(256)
void softmax_den(const int* __restrict__ dst, const unsigned* __restrict__ m,
                 float* __restrict__ sbuf, float* __restrict__ den) {
  int t = blockIdx.x * blockDim.x + threadIdx.x;
  if (t >= N_EDGES * NH) return;
  int e = t >> 3, hh = t & 7;
  int d = dst[e];
  float ex = expf(sbuf[t] - dec_f32(m[d * NH + hh]));
  sbuf[t] = ex;
  atomicAdd(&den[d * NH + hh], ex);
}

// ---------------- 6) h_new[dst] += Wh[src] * a  (one block per edge) -------
__global__ __launch_bounds__(256)
void aggregate(const int* __restrict__ src, const int* __restrict__ dst,
               const float* __restrict__ Wh, const float* __restrict__ sbuf,
               const float* __restrict__ den, float* __restrict__ out) {
  int e  = blockIdx.x;
  int c  = threadIdx.x;                 // 0..255, coalesced over the row
  int hh = c >> 5;
  int s = src[e], d = dst[e];
  float a = sbuf[(size_t)e * NH + hh] / den[d * NH + hh];
  float val = Wh[(size_t)s * HF + c] * a;
  atomicAdd(&out[(size_t)d * HF + c], val);
}

// ---------------- 7) ELU in place on d_out ---------------------------------
__global__ __launch_bounds__(256)
void elu_inplace(float* __restrict__ out, int n) {
  int t = blockIdx.x * blockDim.x + threadIdx.x;
  if (t >= n) return;
  float x = out[t];
  out[t] = x > 0.f ? x : expm1f(x);
}

extern "C" void kernel_launch(void* const* d_in, const int* in_sizes, int n_in,
                              void* d_out, int out_size, void* d_ws, size_t ws_size,
                              hipStream_t stream) {
  const float* h   = (const float*)d_in[0];
  const float* ef  = (const float*)d_in[1];
  const int*   src = (const int*)  d_in[2];
  const int*   dst = (const int*)  d_in[3];
  const float* W_w = (const float*)d_in[4];
  const float* We_w= (const float*)d_in[5];
  const float* a_s = (const float*)d_in[6];
  const float* a_d = (const float*)d_in[7];
  const float* a_e = (const float*)d_in[8];
  float* out = (float*)d_out;

  char* ws = (char*)d_ws;
  const size_t WH_BYTES   = (size_t)N_NODES * HF * 4;   // 51.2 MB
  const size_t SBUF_BYTES = (size_t)N_EDGES * NH * 4;   // 25.6 MB
  const size_t NH_BYTES   = (size_t)N_NODES * NH * 4;   //  1.6 MB
  float*    Wh   = (float*)   (ws);
  float*    sbuf = (float*)   (ws + WH_BYTES);
  float*    ssrc = (float*)   (ws + WH_BYTES + SBUF_BYTES);
  float*    sdst = (float*)   (ws + WH_BYTES + SBUF_BYTES + NH_BYTES);
  unsigned* mmax = (unsigned*)(ws + WH_BYTES + SBUF_BYTES + 2 * NH_BYTES);
  float*    den  = (float*)   (ws + WH_BYTES + SBUF_BYTES + 3 * NH_BYTES);
  float*    V    = (float*)   (ws + WH_BYTES + SBUF_BYTES + 4 * NH_BYTES);

  // zero accumulators (enc(-inf) < enc(any finite) >= 0x007fffff > 0, so 0 is ok for max)
  hipMemsetAsync(out,  0, (size_t)N_NODES * HF * 4, stream);
  hipMemsetAsync(mmax, 0, NH_BYTES, stream);
  hipMemsetAsync(den,  0, NH_BYTES, stream);

  gemm_wh_wmma<<<N_NODES / 16, 256, 0, stream>>>(h, W_w, Wh);
  fold_we<<<1, 512, 0, stream>>>(We_w, a_e, V);
  node_scores<<<(N_NODES * NH + 255) / 256, 256, 0, stream>>>(Wh, a_s, a_d, ssrc, sdst);
  edge_logits<<<(N_EDGES * NH + 255) / 256, 256, 0, stream>>>(ef, src, dst, V, ssrc, sdst, sbuf, mmax);
  softmax_den<<<(N_EDGES * NH + 255) / 256, 256, 0, stream>>>(dst, mmax, sbuf, den);
  aggregate<<<N_EDGES, 256, 0, stream>>>(src, dst, Wh, sbuf, den, out);
  elu_inplace<<<(N_NODES * HF + 255) / 256, 256, 0, stream>>>(out, N_NODES * HF);
}